// Noise_3848290697228
// MI455X (gfx1250) — compile-verified
//
#include <hip/hip_runtime.h>

typedef __attribute__((ext_vector_type(2))) float v2f;
typedef __attribute__((ext_vector_type(8))) float v8f;
typedef unsigned int u32x4 __attribute__((ext_vector_type(4)));
typedef int i32x8 __attribute__((ext_vector_type(8)));
typedef int i32x4 __attribute__((ext_vector_type(4)));

#define SRN    16000
#define HOP    512
#define FRAME  1024
#define KDIM   8001          // SR/2+1
#define KPAD   8064          // K padded to multiple of 64 (zero-filled tail), 126 chunks
#define KCHUNK 64
#define NCHUNK (KPAD / KCHUNK)   // 126
#define MROWS  1000          // B*S
#define MPAD   1024          // M padded (zero rows)
#define NPAD   2048          // 2047 needed ir samples, padded
#define NIRV   2047
#define OUTN   256000        // HOP*S per batch

#define LDS_ROW   66         // 64 data dwords + 2 TDM pad dwords (keeps b64 LDS reads 8B-aligned)
#define TILE_FL   (64 * LDS_ROW)   // floats per staged tile (4224)

// ---- workspace layout (float offsets) ----
#define APAD_OFF 0ull
#define CT_OFF   (APAD_OFF + (size_t)MPAD * KPAD)            //  8,257,536
#define D_OFF    (CT_OFF   + (size_t)NPAD * KPAD)            // 24,772,608
#define W_OFF    (D_OFF    + (size_t)MPAD * NPAD)            // 26,869,760
// total ~27.9M floats ~= 112 MB

// ---- kernel 0a: zero-pad bands into Apad[MPAD][KPAD] ----
__global__ void pad_bands_kernel(const float* __restrict__ bands,
                                 float* __restrict__ Apad) {
    size_t i = (size_t)blockIdx.x * blockDim.x + threadIdx.x;
    size_t total = (size_t)MPAD * KPAD;
    if (i >= total) return;
    int m = (int)(i / KPAD);
    int k = (int)(i % KPAD);
    float v = 0.0f;
    if (m < MROWS && k < KDIM) v = bands[(size_t)m * KDIM + k];
    Apad[i] = v;
}

// ---- kernel 0b: cosine table Ct[n][k] = g_k * cos(2*pi*k*((n-1024) mod 16000)/16000)
__global__ void build_ct_kernel(float* __restrict__ Ct) {
    size_t i = (size_t)blockIdx.x * blockDim.x + threadIdx.x;
    size_t total = (size_t)NPAD * KPAD;
    if (i >= total) return;
    int n = (int)(i / KPAD);
    int k = (int)(i % KPAD);
    float v = 0.0f;
    if (k < KDIM) {
        int j  = (n + (SRN - FRAME)) % SRN;       // (n - 1024) mod 16000
        int ph = (k * j) % SRN;                   // < 2^27, exact in int
        float ang = (float)ph * 3.9269908169872414e-4f;  // 2*pi/16000
        float gk  = (k == 0 || k == KDIM - 1) ? (1.0f / SRN) : (2.0f / SRN);
        v = gk * cosf(ang);
    }
    Ct[i] = v;
}

// ---- TDM: load a 64x64 f32 tile (row stride `row_stride` elems) into LDS at lds_off.
// LDS padding: +2 dwords every 64 dwords -> LDS row stride = 66 dwords.
__device__ __forceinline__ void tdm_load_tile64(unsigned lds_off,
                                                const float* gsrc,
                                                unsigned row_stride) {
    unsigned long long ga = (unsigned long long)(uintptr_t)gsrc;
    u32x4 g0;
    g0.x = 1u;                                         // count=1 (valid descriptor)
    g0.y = lds_off;                                    // lds_addr (bytes)
    g0.z = (unsigned)ga;                               // global_addr[31:0]
    g0.w = ((unsigned)(ga >> 32) & 0x1FFFFFFu)         // global_addr[56:32]
         | 0x80000000u;                                // type = 2 (bits 127:126 = 10b)
    i32x8 g1;
    g1[0] = 0x03520000;        // data_size=4B | pad_enable | pad_interval=64dw | pad_amount=2dw
    g1[1] = (64 << 16);        // tensor_dim0[15:0] = 64 (bits 63:48)
    g1[2] = (64 << 16);        // tensor_dim0[31:16]=0 ; tensor_dim1[15:0] = 64
    g1[3] = (64 << 16);        // tensor_dim1[31:16]=0 ; tile_dim0 = 64
    g1[4] = 64;                // tile_dim1 = 64 ; tile_dim2 = 0
    g1[5] = (int)row_stride;   // tensor_dim0_stride[31:0]
    g1[6] = 0;                 // stride hi / tensor_dim1_stride lo
    g1[7] = 0;
    i32x4 gz4 = {0, 0, 0, 0};            // groups 2/3: dims/tiles unused (<=2D)
    i32x8 gz8 = {0, 0, 0, 0, 0, 0, 0, 0};
    __builtin_amdgcn_tensor_load_to_lds(g0, g1, gz4, gz4, gz8, 0);
}

// ---- kernel 1: D[MPAD][NPAD] = Apad x Ct^T.
// 128-thread block (4 waves) computes a 64x64 output block; A/B 64x64 tiles are
// double-buffered in LDS via the Tensor Data Mover; WMMA f32 16x16x4 from LDS frags.
__global__ __launch_bounds__(128)
void gemm_wmma_kernel(const float* __restrict__ Apad,
                      const float* __restrict__ Ct,
                      float* __restrict__ D) {
    __shared__ __align__(16) float As[2 * TILE_FL];
    __shared__ __align__(16) float Bs[2 * TILE_FL];

    const int tid  = threadIdx.x;
    const int w    = tid >> 5;        // wave 0..3
    const int lane = tid & 31;
    const int lo = lane & 15;
    const int hi = lane >> 4;
    const int kk = hi * 2;            // ISA 16x4 f32 A/B layout: lanes16-31 hold K+2,K+3

    const int mblk = blockIdx.y * 64; // output rows of this block
    const int nblk = blockIdx.x * 64; // output cols of this block
    const int wr = (w >> 1) * 32;     // wave's row offset within block
    const int wc = (w & 1) * 32;      // wave's col offset within block

    const float* gA = Apad + (size_t)mblk * KPAD;
    const float* gB = Ct   + (size_t)nblk * KPAD;
    const unsigned asBase = (unsigned)(uintptr_t)&As[0];
    const unsigned bsBase = (unsigned)(uintptr_t)&Bs[0];
    const unsigned bufBytes = TILE_FL * 4u;

    // LDS fragment row bases (dwords)
    const int rA0 = (wr + lo) * LDS_ROW + kk;
    const int rA1 = rA0 + 16 * LDS_ROW;
    const int rB0 = (wc + lo) * LDS_ROW + kk;
    const int rB1 = rB0 + 16 * LDS_ROW;

    v8f acc00 = {}, acc01 = {}, acc10 = {}, acc11 = {};

    if (w == 0) {                      // prologue: DMA chunk 0 into buffer 0
        tdm_load_tile64(asBase, gA, KPAD);
        tdm_load_tile64(bsBase, gB, KPAD);
    }

    for (int c = 0; c < NCHUNK; ++c) {
        const int buf = c & 1;
        if (w == 0) __builtin_amdgcn_s_wait_tensorcnt(0);   // chunk c landed in LDS
        __syncthreads();                                    // visible to all 4 waves
        if (w == 0 && (c + 1) < NCHUNK) {                   // prefetch chunk c+1
            const int nb = (c + 1) & 1;
            tdm_load_tile64(asBase + nb * bufBytes, gA + (size_t)(c + 1) * KCHUNK, KPAD);
            tdm_load_tile64(bsBase + nb * bufBytes, gB + (size_t)(c + 1) * KCHUNK, KPAD);
        }

        const float* as = &As[buf * TILE_FL];
        const float* bs = &Bs[buf * TILE_FL];
#pragma unroll
        for (int ks = 0; ks < KCHUNK / 4; ++ks) {
            const int kc = ks * 4;
            v2f a0 = *(const v2f*)(as + rA0 + kc);
            v2f a1 = *(const v2f*)(as + rA1 + kc);
            v2f b0 = *(const v2f*)(bs + rB0 + kc);
            v2f b1 = *(const v2f*)(bs + rB1 + kc);
            acc00 = __builtin_amdgcn_wmma_f32_16x16x4_f32(false, a0, false, b0,
                                                          (short)0, acc00, false, false);
            acc01 = __builtin_amdgcn_wmma_f32_16x16x4_f32(false, a0, false, b1,
                                                          (short)0, acc01, false, false);
            acc10 = __builtin_amdgcn_wmma_f32_16x16x4_f32(false, a1, false, b0,
                                                          (short)0, acc10, false, false);
            acc11 = __builtin_amdgcn_wmma_f32_16x16x4_f32(false, a1, false, b1,
                                                          (short)0, acc11, false, false);
        }
        __syncthreads();               // everyone done reading buf before TDM reuses it
    }

    // 16x16 f32 C/D layout: VGPR v holds M = v + 8*hi, N = lo
#pragma unroll
    for (int v = 0; v < 8; ++v) {
        int r0 = mblk + wr + v + 8 * hi;
        int r1 = r0 + 16;
        int c0 = nblk + wc + lo;
        int c1 = c0 + 16;
        D[(size_t)r0 * NPAD + c0] = acc00[v];
        D[(size_t)r0 * NPAD + c1] = acc01[v];
        D[(size_t)r1 * NPAD + c0] = acc10[v];
        D[(size_t)r1 * NPAD + c1] = acc11[v];
    }
}

// ---- kernel 2: per-(b,s) correlation with the 2047-tap ir window + Hann window ----
// filtered[t] = sum_m frames[m] * irv[m - t + 1023];  W[bs][t] = filtered[t]*hann[t]
__global__ __launch_bounds__(256)
void corr_window_kernel(const float* __restrict__ noise,
                        const float* __restrict__ D,
                        float* __restrict__ W) {
    __shared__ float fr[FRAME];
    __shared__ float irv[NIRV + 1];

    const int bs = blockIdx.x;            // 0..999
    const int b  = bs / 500;
    const int s  = bs % 500;
    const float* nz = noise + (size_t)b * (500 * HOP + HOP) + (size_t)s * HOP;

    for (int m = threadIdx.x; m < FRAME; m += 256) fr[m] = 2.0f * nz[m] - 1.0f;
    const float* dd = D + (size_t)bs * NPAD;
    for (int j = threadIdx.x; j < NIRV; j += 256) irv[j] = dd[j];
    __syncthreads();

    const int t0 = threadIdx.x;
    float acc0 = 0.f, acc1 = 0.f, acc2 = 0.f, acc3 = 0.f;
    for (int m = 0; m < FRAME; ++m) {
        float f = fr[m];
        int base = m - t0 + (FRAME - 1);  // in [768, 2046] for acc0
        acc0 += f * irv[base];
        acc1 += f * irv[base - 256];
        acc2 += f * irv[base - 512];
        acc3 += f * irv[base - 768];
    }
    float accs[4] = {acc0, acc1, acc2, acc3};
#pragma unroll
    for (int r = 0; r < 4; ++r) {
        int t = t0 + 256 * r;
        float win = 0.5f - 0.5f * cosf(6.2831853071795865e0f * (float)t / (float)FRAME);
        W[(size_t)bs * FRAME + t] = accs[r] * win;
    }
}

// ---- kernel 3: deterministic overlap-add gather (no float atomics) ----
__global__ void ola_gather_kernel(const float* __restrict__ W,
                                  float* __restrict__ out) {
    int i = blockIdx.x * blockDim.x + threadIdx.x;
    if (i >= 2 * OUTN) return;
    int b = i / OUTN;
    int n = i % OUTN;
    int s0 = n >> 9;          // n / 512
    int t0 = n & 511;
    float v = W[(size_t)(b * 500 + s0) * FRAME + t0];
    if (s0 > 0) v += W[(size_t)(b * 500 + s0 - 1) * FRAME + t0 + 512];
    out[i] = v;
}

extern "C" void kernel_launch(void* const* d_in, const int* in_sizes, int n_in,
                              void* d_out, int out_size, void* d_ws, size_t ws_size,
                              hipStream_t stream) {
    const float* bands = (const float*)d_in[0];   // (2,500,8001) f32
    const float* noise = (const float*)d_in[1];   // (2,256512)   f32
    float* out = (float*)d_out;                   // (2,1,256000) f32
    float* ws  = (float*)d_ws;

    float* Apad = ws + APAD_OFF;
    float* Ct   = ws + CT_OFF;
    float* D    = ws + D_OFF;
    float* W    = ws + W_OFF;

    {   // 0a: pad bands
        size_t total = (size_t)MPAD * KPAD;
        pad_bands_kernel<<<dim3((unsigned)((total + 255) / 256)), dim3(256), 0, stream>>>(bands, Apad);
    }
    {   // 0b: cosine table
        size_t total = (size_t)NPAD * KPAD;
        build_ct_kernel<<<dim3((unsigned)((total + 255) / 256)), dim3(256), 0, stream>>>(Ct);
    }
    {   // 1: WMMA GEMM with TDM-staged LDS double buffering
        gemm_wmma_kernel<<<dim3(NPAD / 64, MPAD / 64), dim3(128), 0, stream>>>(Apad, Ct, D);
    }
    {   // 2: correlation + window
        corr_window_kernel<<<dim3(MROWS), dim3(256), 0, stream>>>(noise, D, W);
    }
    {   // 3: overlap-add gather
        ola_gather_kernel<<<dim3((2 * OUTN + 255) / 256), dim3(256), 0, stream>>>(W, out);
    }
}